// HeteroMCGCN_7576322310515
// MI455X (gfx1250) — compile-verified
//
#include <hip/hip_runtime.h>
#include <hip/hip_bf16.h>
#include <math.h>

// ---------------------------------------------------------------------------
// HeteroMCGCN on gfx1250 (MI455X).  fp32 end-to-end; GEMMs via
// V_WMMA_F32_16X16X4_F32 with 2x4 register blocking (32x64 tile per wave32,
// 8 accumulators) so each K-step issues 8 WMMAs against 10 VMEM loads.
// Edge embedding is rank-1 in the scalar edge value, so the per-layer
// bias/gate reduce to edge * w1[h] and edge * w2[h], w12 = We_ch . We[layer].
// ---------------------------------------------------------------------------

#define HID   128
#define HEADS 8
#define HD    16
#define LYR   3
#define BB    8
#define NN    200
#define ROWS  (BB * NN)     // 1600 = 50 tiles of 32 rows
#define FF    (4 * HID)     // 512
#define TOT   (3 * HID)     // 384
#define FFT   (4 * TOT)     // 1536

typedef __attribute__((ext_vector_type(2))) float v2f;
typedef __attribute__((ext_vector_type(8))) float v8f;

static __device__ __forceinline__ float wave_sum(float v) {
    for (int o = 16; o; o >>= 1) v += __shfl_xor(v, o, 32);
    return v;
}
static __device__ __forceinline__ float wave_max(float v) {
    for (int o = 16; o; o >>= 1) v = fmaxf(v, __shfl_xor(v, o, 32));
    return v;
}

// h0[row, c] = nf[row,0]*Wn[0,c] + nf[row,1]*Wn[1,c]
__global__ void h0_kernel(const float* __restrict__ nf, const float* __restrict__ Wn,
                          float* __restrict__ h0) {
    int row = blockIdx.x, c = threadIdx.x;
    h0[(size_t)row * HID + c] =
        nf[row * 2 + 0] * Wn[c] + nf[row * 2 + 1] * Wn[HID + c];
}

// w12[ch*48 + i*16 + j] = sum_k We_ch[k] * We[i][k][j]   (2*HEADS=16 per layer)
__global__ void w12_kernel(const float* __restrict__ wef, const float* __restrict__ wed,
                           const float* __restrict__ Wf,  const float* __restrict__ Wd,
                           float* __restrict__ w12) {
    int t = threadIdx.x;                 // 0..95
    int ch = t / 48, r = t % 48, i = r / 16, j = r % 16;
    const float* wch = ch ? wed : wef;
    const float* W   = ch ? Wd  : Wf;
    float s = 0.f;
    for (int k = 0; k < HID; ++k)
        s += wch[k] * W[((size_t)i * HID + k) * 16 + j];
    w12[t] = s;
}

// dst[r*ds + coff + c] = src[r*ss + c],  c in [0,128)
__global__ void copy_cols(const float* __restrict__ src, int ss,
                          float* __restrict__ dst, int ds, int coff) {
    int r = blockIdx.x, c = threadIdx.x;
    dst[(size_t)r * ds + coff + c] = src[(size_t)r * ss + c];
}

// out = LN(in + add) * sc + bi   over 128 features, one wave per row
__global__ void ln_kernel(const float* __restrict__ in, const float* __restrict__ addv,
                          const float* __restrict__ sc, const float* __restrict__ bi,
                          float* __restrict__ out) {
    int row = blockIdx.x, lane = threadIdx.x;
    const float* ir = in + (size_t)row * HID;
    const float* ar = addv ? addv + (size_t)row * HID : nullptr;
    float x[4];
#pragma unroll
    for (int j = 0; j < 4; ++j) {
        int c = lane + 32 * j;
        x[j] = ir[c] + (ar ? ar[c] : 0.f);
    }
    float m = wave_sum(x[0] + x[1] + x[2] + x[3]) * (1.f / 128.f);
    float v = 0.f;
#pragma unroll
    for (int j = 0; j < 4; ++j) { float d = x[j] - m; v += d * d; }
    v = wave_sum(v) * (1.f / 128.f);
    float rs = rsqrtf(v + 1e-5f);
#pragma unroll
    for (int j = 0; j < 4; ++j) {
        int c = lane + 32 * j;
        out[(size_t)row * HID + c] = (x[j] - m) * rs * sc[c] + bi[c];
    }
}

// C[M,N] = act(A[M,K] @ W[K,N]) (+ add1 + add2).
// One wave32 per 32x64 output tile: 2 M-tiles x 4 N-tiles of 16x16,
// 8 fp32-WMMA accumulators.  grid = (N/64, M/32), block = 32.
// Requires M % 32 == 0, N % 64 == 0, K % 4 == 0 (true for all calls here).
__global__ void gemm_wmma(const float* __restrict__ A, const float* __restrict__ W,
                          float* __restrict__ C,
                          const float* __restrict__ add1, const float* __restrict__ add2,
                          int K, int N, int ldc, int coff, int relu) {
    int lane = threadIdx.x;
    int r  = lane & 15;
    int kh = (lane >> 4) * 2;                 // 0 or 2
    int tn0 = blockIdx.x * 64;
    int tm0 = blockIdx.y * 32;

    const float* Ap0 = A + (size_t)(tm0 + r) * K + kh;
    const float* Ap1 = A + (size_t)(tm0 + 16 + r) * K + kh;
    const float* Wp0 = W + (size_t)kh * N + (tn0 + r);
    const float* Wp1 = Wp0 + 16;
    const float* Wp2 = Wp0 + 32;
    const float* Wp3 = Wp0 + 48;

    v8f acc[2][4];
#pragma unroll
    for (int i = 0; i < 2; ++i)
#pragma unroll
        for (int t = 0; t < 4; ++t)
            acc[i][t] = (v8f){0.f, 0.f, 0.f, 0.f, 0.f, 0.f, 0.f, 0.f};

    for (int k = 0; k < K; k += 4) {
        v2f a0, a1, b0, b1, b2, b3;
        a0.x = Ap0[k];            a0.y = Ap0[k + 1];
        a1.x = Ap1[k];            a1.y = Ap1[k + 1];
        size_t ro0 = (size_t)k * N, ro1 = (size_t)(k + 1) * N;
        b0.x = Wp0[ro0];          b0.y = Wp0[ro1];
        b1.x = Wp1[ro0];          b1.y = Wp1[ro1];
        b2.x = Wp2[ro0];          b2.y = Wp2[ro1];
        b3.x = Wp3[ro0];          b3.y = Wp3[ro1];
        acc[0][0] = __builtin_amdgcn_wmma_f32_16x16x4_f32(false, a0, false, b0, (short)0, acc[0][0], false, false);
        acc[0][1] = __builtin_amdgcn_wmma_f32_16x16x4_f32(false, a0, false, b1, (short)0, acc[0][1], false, false);
        acc[0][2] = __builtin_amdgcn_wmma_f32_16x16x4_f32(false, a0, false, b2, (short)0, acc[0][2], false, false);
        acc[0][3] = __builtin_amdgcn_wmma_f32_16x16x4_f32(false, a0, false, b3, (short)0, acc[0][3], false, false);
        acc[1][0] = __builtin_amdgcn_wmma_f32_16x16x4_f32(false, a1, false, b0, (short)0, acc[1][0], false, false);
        acc[1][1] = __builtin_amdgcn_wmma_f32_16x16x4_f32(false, a1, false, b1, (short)0, acc[1][1], false, false);
        acc[1][2] = __builtin_amdgcn_wmma_f32_16x16x4_f32(false, a1, false, b2, (short)0, acc[1][2], false, false);
        acc[1][3] = __builtin_amdgcn_wmma_f32_16x16x4_f32(false, a1, false, b3, (short)0, acc[1][3], false, false);
    }

    int rsub = 8 * (lane >> 4);               // C/D: row = r + 8*(lane/16)
#pragma unroll
    for (int i = 0; i < 2; ++i) {
        int rbase = tm0 + i * 16 + rsub;
#pragma unroll
        for (int t = 0; t < 4; ++t) {
            int col = tn0 + t * 16 + r;
#pragma unroll
            for (int rr = 0; rr < 8; ++rr) {
                float val = acc[i][t][rr];
                if (relu) val = fmaxf(val, 0.f);
                size_t o = (size_t)(rbase + rr) * ldc + coff + col;
                if (add1) val += add1[o];
                if (add2) val += add2[o];
                C[o] = val;
            }
        }
    }
}

// One wave computes one (b, head, query-row) of attention.
// score = q.k/4 (+ edge*w1); softmax; gate *= edge*w2 (if edge); y = att @ v.
__global__ void attn_kernel(const float* __restrict__ qp, const float* __restrict__ kp,
                            const float* __restrict__ vp,
                            int qs, int ks, int vs,
                            const float* __restrict__ edge, const float* __restrict__ w12,
                            float* __restrict__ out, int os, int ooff) {
    int b = blockIdx.z, h = blockIdx.y, i = blockIdx.x;
    int lane = threadIdx.x;
    bool useE = (edge != nullptr);
    float w1 = useE ? w12[h] : 0.f;
    float w2 = useE ? w12[HEADS + h] : 0.f;

    float qv[HD];
    const float* qrow = qp + (size_t)(b * NN + i) * qs + h * HD;
#pragma unroll
    for (int d = 0; d < HD; ++d) qv[d] = qrow[d];
    const float* erow = useE ? edge + (size_t)(b * NN + i) * NN : nullptr;

    float sj[7];
    float mx = -1e30f;
    int cnt = 0;
    for (int j = lane; j < NN; j += 32) {
        const float* krow = kp + (size_t)(b * NN + j) * ks + h * HD;
        float s = 0.f;
#pragma unroll
        for (int d = 0; d < HD; ++d) s += qv[d] * krow[d];
        s *= 0.25f;                                // 1/sqrt(16)
        if (useE) s += erow[j] * w1;
        sj[cnt++] = s;
        mx = fmaxf(mx, s);
    }
    mx = wave_max(mx);

    float se = 0.f;
    cnt = 0;
    for (int j = lane; j < NN; j += 32) {
        float p = expf(sj[cnt] - mx);
        sj[cnt++] = p;
        se += p;
    }
    se = wave_sum(se);
    float inv = 1.f / se;

    float ya[HD];
#pragma unroll
    for (int d = 0; d < HD; ++d) ya[d] = 0.f;
    cnt = 0;
    for (int j = lane; j < NN; j += 32) {
        float g = sj[cnt++] * inv;
        if (useE) g *= erow[j] * w2;
        const float* vrow = vp + (size_t)(b * NN + j) * vs + h * HD;
#pragma unroll
        for (int d = 0; d < HD; ++d) ya[d] += g * vrow[d];
    }
#pragma unroll
    for (int d = 0; d < HD; ++d) ya[d] = wave_sum(ya[d]);

    if (lane == 0) {
        float* orow = out + (size_t)(b * NN + i) * os + ooff + h * HD;
#pragma unroll
        for (int d = 0; d < HD; ++d) orow[d] = ya[d];
    }
}

// out[row] = 10*tanh((fin[row,:] . dec) / sqrt(128))
__global__ void decode_kernel(const float* __restrict__ fin, const float* __restrict__ dec,
                              float* __restrict__ out) {
    int row = blockIdx.x, lane = threadIdx.x;
    float s = 0.f;
    for (int c = lane; c < TOT; c += 32) s += fin[(size_t)row * TOT + c] * dec[c];
    s = wave_sum(s);
    if (lane == 0) out[row] = 10.f * tanhf(s * 0.08838834764831845f);
}

extern "C" void kernel_launch(void* const* d_in, const int* in_sizes, int n_in,
                              void* d_out, int out_size, void* d_ws, size_t ws_size,
                              hipStream_t stream) {
    (void)in_sizes; (void)n_in; (void)out_size; (void)ws_size;
    const float* nf     = (const float*)d_in[0];
    const float* eflow  = (const float*)d_in[1];
    const float* edist  = (const float*)d_in[2];
    const float* Wnode  = (const float*)d_in[3];
    const float* Weflow = (const float*)d_in[4];
    const float* Wedist = (const float*)d_in[5];
    struct P { const float *ln1s, *ln1b, *ln2s, *ln2b, *Wh, *We, *mW1, *mW2; };
    P prm[2];
    for (int c = 0; c < 2; ++c) {
        int b = 6 + 8 * c;   // nested dict flattened in insertion order
        prm[c].ln1s = (const float*)d_in[b + 0];
        prm[c].ln1b = (const float*)d_in[b + 1];
        prm[c].ln2s = (const float*)d_in[b + 2];
        prm[c].ln2b = (const float*)d_in[b + 3];
        prm[c].Wh   = (const float*)d_in[b + 4];
        prm[c].We   = (const float*)d_in[b + 5];
        prm[c].mW1  = (const float*)d_in[b + 6];
        prm[c].mW2  = (const float*)d_in[b + 7];
    }
    const float* mlpW1 = (const float*)d_in[22];
    const float* mlpW2 = (const float*)d_in[23];
    const float* decW  = (const float*)d_in[24];
    float* out = (float*)d_out;

    // workspace carving (floats)
    float* ws = (float*)d_ws;
    size_t off = 0;
    auto carve = [&](size_t n) { float* p = ws + off; off += n; return p; };
    float* h0    = carve((size_t)ROWS * HID);
    float* chA[2] = { carve((size_t)ROWS * HID), carve((size_t)ROWS * HID) };
    float* chB[2] = { carve((size_t)ROWS * HID), carve((size_t)ROWS * HID) };
    float* hn    = carve((size_t)ROWS * HID);
    float* qkv   = carve((size_t)ROWS * 3 * HID);
    float* ybuf  = carve((size_t)ROWS * HID);
    float* zbuf  = carve((size_t)ROWS * HID);
    float* mid   = carve((size_t)ROWS * FF);
    float* hcat  = carve((size_t)ROWS * TOT);
    float* fmid  = carve((size_t)ROWS * FFT);
    float* ffout = carve((size_t)ROWS * TOT);
    float* w12   = carve(96);

    h0_kernel<<<ROWS, HID, 0, stream>>>(nf, Wnode, h0);
    w12_kernel<<<1, 96, 0, stream>>>(Weflow, Wedist, prm[0].We, prm[1].We, w12);

    const float* edges[2] = { eflow, edist };
    float* hfin[2];
    for (int c = 0; c < 2; ++c) {
        float* cur = chA[c];
        float* nxt = chB[c];
        copy_cols<<<ROWS, HID, 0, stream>>>(h0, HID, cur, HID, 0);
        for (int i = 0; i < LYR; ++i) {
            ln_kernel<<<ROWS, 32, 0, stream>>>(cur, nullptr,
                prm[c].ln1s + i * HID, prm[c].ln1b + i * HID, hn);
            gemm_wmma<<<dim3(3 * HID / 64, ROWS / 32), 32, 0, stream>>>(
                hn, prm[c].Wh + (size_t)i * HID * 3 * HID, qkv,
                nullptr, nullptr, HID, 3 * HID, 3 * HID, 0, 0);
            attn_kernel<<<dim3(NN, HEADS, BB), 32, 0, stream>>>(
                qkv, qkv + HID, qkv + 2 * HID, 3 * HID, 3 * HID, 3 * HID,
                edges[c], w12 + c * 48 + i * 16, ybuf, HID, 0);
            ln_kernel<<<ROWS, 32, 0, stream>>>(ybuf, cur,
                prm[c].ln2s + i * HID, prm[c].ln2b + i * HID, zbuf);
            gemm_wmma<<<dim3(FF / 64, ROWS / 32), 32, 0, stream>>>(
                zbuf, prm[c].mW1 + (size_t)i * HID * FF, mid,
                nullptr, nullptr, HID, FF, FF, 0, 1);
            // h_{i+1} = mlp_out + y + h_i   (res == h at every layer entry)
            gemm_wmma<<<dim3(HID / 64, ROWS / 32), 32, 0, stream>>>(
                mid, prm[c].mW2 + (size_t)i * FF * HID, nxt,
                ybuf, cur, FF, HID, HID, 0, 0);
            float* t = cur; cur = nxt; nxt = t;
        }
        hfin[c] = cur;
    }

    // h_cat = [h0 | cross_att(flow->dist) | cross_att(dist->flow)]
    copy_cols<<<ROWS, HID, 0, stream>>>(h0, HID, hcat, TOT, 0);
    attn_kernel<<<dim3(NN, HEADS, BB), 32, 0, stream>>>(
        hfin[0], hfin[1], hfin[1], HID, HID, HID, nullptr, nullptr, hcat, TOT, HID);
    attn_kernel<<<dim3(NN, HEADS, BB), 32, 0, stream>>>(
        hfin[1], hfin[0], hfin[0], HID, HID, HID, nullptr, nullptr, hcat, TOT, 2 * HID);

    gemm_wmma<<<dim3(FFT / 64, ROWS / 32), 32, 0, stream>>>(
        hcat, mlpW1, fmid, nullptr, nullptr, TOT, FFT, FFT, 0, 1);
    gemm_wmma<<<dim3(TOT / 64, ROWS / 32), 32, 0, stream>>>(
        fmid, mlpW2, ffout, nullptr, nullptr, FFT, TOT, TOT, 0, 0);
    decode_kernel<<<ROWS, 32, 0, stream>>>(ffout, decW, out);
}